// INCODE_16922171146384
// MI455X (gfx1250) — compile-verified
//
#include <hip/hip_runtime.h>
#include <hip/hip_bf16.h>
#include <math.h>

typedef float v2f __attribute__((ext_vector_type(2)));
typedef float v8f __attribute__((ext_vector_type(8)));

#define NLAYERS   256
#define LW_STRIDE 20          // 16 prescaled W + 4 prescaled bias floats per layer

// Full half-wave swap: lane i <-> lane i^16  (ds_swizzle group-of-32, xor=0x10, and=0x1f)
__device__ __forceinline__ float swapx16(float x) {
    return __int_as_float(__builtin_amdgcn_ds_swizzle(__float_as_int(x), 0x401F));
}

__global__ __launch_bounds__(256)
void incode_siren_kernel(const float* __restrict__ coords,
                         const float* __restrict__ w_first,
                         const float* __restrict__ b_first,
                         const float* __restrict__ w_hidden,
                         const float* __restrict__ b_hidden,
                         const float* __restrict__ w_out,
                         const float* __restrict__ b_out,
                         const float* __restrict__ a_param,
                         const float* __restrict__ b_param,
                         const float* __restrict__ c_param,
                         const float* __restrict__ d_param,
                         float* __restrict__ out, int n)
{
    __shared__ float lw[NLAYERS * LW_STRIDE];   // 20 KB

    const float INV2PI = 0.15915494309189535f;  // v_sin_f32 computes sin(2*pi*x)
    const float amp  = __expf(a_param[0]);
    const float freq = __expf(b_param[0]);
    const float cpar = c_param[0];
    const float dpar = d_param[0];
    const float s_hid  = freq * 30.0f * INV2PI;
    const float c_norm = cpar * INV2PI;

    // ---- stage pre-scaled hidden weights into LDS: layer l = threadIdx.x ----
    {
        const int l = threadIdx.x;              // blockDim.x == 256 == NLAYERS
        const float* wsrc = w_hidden + l * 16;
        const float* bsrc = b_hidden + l * 4;
        float* dst = lw + l * LW_STRIDE;
        #pragma unroll
        for (int e = 0; e < 16; ++e) dst[e] = s_hid * wsrc[e];
        #pragma unroll
        for (int e = 0; e < 4; ++e)  dst[16 + e] = fmaf(s_hid, bsrc[e], c_norm);
    }
    __syncthreads();

    const int lane  = threadIdx.x & 31;
    const int wave  = threadIdx.x >> 5;
    const int nl    = lane & 15;     // sample-within-tile (B/D column)
    const int khalf = lane >> 4;     // lane half selects K rows {0,1} vs {2,3}
    const int k0 = khalf * 2;
    const int k1 = k0 + 1;

    const int base = blockIdx.x * 256 + wave * 32;   // 2 tiles x 16 samples per wave
    const int s0 = base + nl;
    const int s1 = base + 16 + nl;
    const int s0c = (s0 < n) ? s0 : (n - 1);
    const int s1c = (s1 < n) ? s1 : (n - 1);

    // ---- first sine layer (omega0 = 3000), directly in B layout ----
    const float f1   = freq * 3000.0f * INV2PI;
    const float wf0 = w_first[k0], wf1 = w_first[k1];
    const float bf0 = b_first[k0], bf1 = b_first[k1];
    const float co0 = coords[s0c];
    const float co1 = coords[s1c];

    float b0A, b1A, b0B, b1B;
    {
        float t;
        t = fmaf(f1, fmaf(co0, wf0, bf0), c_norm); t -= rintf(t);
        b0A = fmaf(amp, __builtin_amdgcn_sinf(t), dpar);
        t = fmaf(f1, fmaf(co0, wf1, bf1), c_norm); t -= rintf(t);
        b1A = fmaf(amp, __builtin_amdgcn_sinf(t), dpar);
        t = fmaf(f1, fmaf(co1, wf0, bf0), c_norm); t -= rintf(t);
        b0B = fmaf(amp, __builtin_amdgcn_sinf(t), dpar);
        t = fmaf(f1, fmaf(co1, wf1, bf1), c_norm); t -= rintf(t);
        b1B = fmaf(amp, __builtin_amdgcn_sinf(t), dpar);
    }

    // ---- 256 hidden layers via V_WMMA_F32_16X16X4_F32, chained through B ----
    const v8f zeroC = {0.f, 0.f, 0.f, 0.f, 0.f, 0.f, 0.f, 0.f};
    const bool lo = (lane < 16);
    // A(m,k): lanes hold row (lane&3) duplicated into don't-care rows 4..15,
    // columns k0/k0+1 -> consecutive floats -> single ds_load_b64 per layer.
    const float* wl = lw + ((lane & 3) * 4 + k0);
    const float* bl = lw + (16 + k0);

    for (int l = 0; l < NLAYERS; ++l) {
        const float a0 = wl[0], a1 = wl[1];      // A VGPR0, VGPR1
        const float bb0 = bl[0], bb1 = bl[1];    // per-lane bias for B rows
        wl += LW_STRIDE;  bl += LW_STRIDE;

        v2f A;  A.x = a0;  A.y = a1;
        v2f BA; BA.x = b0A; BA.y = b1A;
        v2f BB; BB.x = b0B; BB.y = b1B;

        v8f dA = __builtin_amdgcn_wmma_f32_16x16x4_f32(false, A, false, BA,
                                                       (short)0, zeroC, false, false);
        v8f dB = __builtin_amdgcn_wmma_f32_16x16x4_f32(false, A, false, BB,
                                                       (short)0, zeroC, false, false);

        // D -> next-layer B repack: rows {0,2} -> b0, rows {1,3} -> b1.
        // Swizzles execute under full EXEC; select is a branchless cndmask.
        const float sw2A = swapx16(dA[2]);
        const float sw3A = swapx16(dA[3]);
        const float sw2B = swapx16(dB[2]);
        const float sw3B = swapx16(dB[3]);
        const float p0A = lo ? dA[0] : sw2A;
        const float p1A = lo ? dA[1] : sw3A;
        const float p0B = lo ? dB[0] : sw2B;
        const float p1B = lo ? dB[1] : sw3B;

        // h = amp * sin(2*pi * (p + b')) + d   (|p+b'| << 1, in-range for v_sin)
        b0A = fmaf(amp, __builtin_amdgcn_sinf(p0A + bb0), dpar);
        b1A = fmaf(amp, __builtin_amdgcn_sinf(p1A + bb1), dpar);
        b0B = fmaf(amp, __builtin_amdgcn_sinf(p0B + bb0), dpar);
        b1B = fmaf(amp, __builtin_amdgcn_sinf(p1B + bb1), dpar);
    }

    // ---- output linear layer: dot over 4 hidden units split across lane halves ----
    const float wo0 = w_out[k0], wo1 = w_out[k1];
    const float bo  = b_out[0];
    float pA = fmaf(b0A, wo0, b1A * wo1);
    float pB = fmaf(b0B, wo0, b1B * wo1);
    pA += swapx16(pA);   // add the other half's (k=2,3) partial
    pB += swapx16(pB);
    if (lane < 16) {
        if (s0 < n) out[s0] = pA + bo;
        if (s1 < n) out[s1] = pB + bo;
    }
}

extern "C" void kernel_launch(void* const* d_in, const int* in_sizes, int n_in,
                              void* d_out, int out_size, void* d_ws, size_t ws_size,
                              hipStream_t stream) {
    const float* coords   = (const float*)d_in[0];
    const float* w_first  = (const float*)d_in[1];
    const float* b_first  = (const float*)d_in[2];
    const float* w_hidden = (const float*)d_in[3];
    const float* b_hidden = (const float*)d_in[4];
    const float* w_out    = (const float*)d_in[5];
    const float* b_out    = (const float*)d_in[6];
    const float* ap = (const float*)d_in[7];
    const float* bp = (const float*)d_in[8];
    const float* cp = (const float*)d_in[9];
    const float* dp = (const float*)d_in[10];
    float* out = (float*)d_out;

    const int n = in_sizes[0];                 // coords is [N,1] -> N
    const int blocks = (n + 255) / 256;        // 256 threads = 8 waves x 32 samples
    hipLaunchKernelGGL(incode_siren_kernel, dim3(blocks), dim3(256), 0, stream,
                       coords, w_first, b_first, w_hidden, b_hidden, w_out, b_out,
                       ap, bp, cp, dp, out, n);
}